// CustomLoss_28827820491181
// MI455X (gfx1250) — compile-verified
//
#include <hip/hip_runtime.h>

// Dead-zone thresholded squared error, mean over N fp32 elements.
// Pass 1: streaming NT b128 loads + FMA accumulate -> per-block partials in ws.
// Pass 2: one wave32 reduces the 2048 partials; final 32->1 cross-lane sum done
//         with a single V_WMMA_F32_16X16X4_F32 (A = partials, B = ones).

#define ERROR_OK_F 0.1f
#define NBLOCKS 2048
#define NTHREADS 256

typedef __attribute__((ext_vector_type(2))) float v2f;
typedef __attribute__((ext_vector_type(4))) float v4f;
typedef __attribute__((ext_vector_type(8))) float v8f;

__global__ void deadzone_partial(const float* __restrict__ a,
                                 const float* __restrict__ b,
                                 float* __restrict__ partials,
                                 long long n) {
    const long long tid    = (long long)blockIdx.x * blockDim.x + threadIdx.x;
    const long long stride = (long long)gridDim.x * blockDim.x;
    const long long nv     = n >> 2;  // float4 count

    const v4f* __restrict__ av = (const v4f*)a;
    const v4f* __restrict__ bv = (const v4f*)b;

    float acc = 0.0f;
#pragma unroll 4
    for (long long v = tid; v < nv; v += stride) {
        v4f x = __builtin_nontemporal_load(av + v);  // global_load_b128 ... nt
        v4f y = __builtin_nontemporal_load(bv + v);
#pragma unroll
        for (int j = 0; j < 4; ++j) {
            float d = x[j] - y[j];
            float z = (__builtin_fabsf(d) < ERROR_OK_F) ? 0.0f : d;
            acc = __builtin_fmaf(z, z, acc);
        }
    }
    // scalar tail (N % 4 != 0); empty for N = 2^25
    for (long long i = (nv << 2) + tid; i < n; i += stride) {
        float d = a[i] - b[i];
        float z = (__builtin_fabsf(d) < ERROR_OK_F) ? 0.0f : d;
        acc = __builtin_fmaf(z, z, acc);
    }

    // wave32 tree reduce
#pragma unroll
    for (int off = 16; off > 0; off >>= 1)
        acc += __shfl_down(acc, off, 32);

    __shared__ float wsum[NTHREADS / 32];
    const int lane = threadIdx.x & 31;
    const int wave = threadIdx.x >> 5;
    if (lane == 0) wsum[wave] = acc;
    __syncthreads();

    if (threadIdx.x == 0) {
        float s = 0.0f;
#pragma unroll
        for (int w = 0; w < NTHREADS / 32; ++w) s += wsum[w];
        partials[blockIdx.x] = s;
    }
}

// One wave (32 threads), EXEC all-1s (uniform control flow) as WMMA requires.
__global__ void deadzone_final(const float* __restrict__ partials,
                               float* __restrict__ out,
                               long long n) {
    float p = 0.0f;
    for (int i = threadIdx.x; i < NBLOCKS; i += 32) p += partials[i];

    // A (16x4 f32): VGPR0 = p  => A[m][0] = p[m] (lanes 0-15), A[m][2] = p[m+16]
    //               VGPR1 = 0  => A[m][1] = A[m][3] = 0
    // B (4x16 f32) = ones  =>  D[m][n] = p[m] + p[m+16]
    v2f amat; amat[0] = p;    amat[1] = 0.0f;
    v2f bmat; bmat[0] = 1.0f; bmat[1] = 1.0f;
    v8f c = {};
    v8f d = __builtin_amdgcn_wmma_f32_16x16x4_f32(
        /*neg_a=*/false, amat, /*neg_b=*/false, bmat,
        /*c_mod=*/(short)0, c, /*reuse_a=*/false, /*reuse_b=*/false);

    float s = 0.0f;
#pragma unroll
    for (int j = 0; j < 8; ++j) s += d[j];  // lane L<16: rows 0-7; lane L>=16: rows 8-15

    float other = __shfl_xor(s, 16, 32);    // pair lane0 <-> lane16
    if (threadIdx.x == 0) out[0] = (s + other) / (float)n;
}

extern "C" void kernel_launch(void* const* d_in, const int* in_sizes, int n_in,
                              void* d_out, int out_size, void* d_ws, size_t ws_size,
                              hipStream_t stream) {
    const float* a = (const float*)d_in[0];
    const float* b = (const float*)d_in[1];
    float* out      = (float*)d_out;
    float* partials = (float*)d_ws;   // NBLOCKS floats = 8 KB scratch
    const long long n = (long long)in_sizes[0];

    deadzone_partial<<<NBLOCKS, NTHREADS, 0, stream>>>(a, b, partials, n);
    deadzone_final<<<1, 32, 0, stream>>>(partials, out, n);
}